// SliceColorShader_24326694765032
// MI455X (gfx1250) — compile-verified
//
#include <hip/hip_runtime.h>

// SliceColorShader: per-pixel argmax-of-3-bary + double gather + masked select.
// Pure memory-bound: ~235 MB streamed/call -> ~10us floor at 23.3 TB/s HBM.
// Gather tables (faces 2.4 MB + verts_colors 1.2 MB) stay L2-resident (192 MB);
// streams use gfx1250 non-temporal hints (th:NT on global_load_b128 /
// global_store_b128) so they don't evict the tables. 4 pixels per thread =>
// all stream traffic is 128-bit aligned. Fast/tail split is BLOCK-uniform so
// the hot path has no exec-mask divergence (pure s_cmp/s_cbranch entry).

typedef int   v4i __attribute__((ext_vector_type(4)));
typedef float v4f __attribute__((ext_vector_type(4)));

struct __attribute__((packed)) f3 { float x, y, z; };   // 12B -> global_load_b96

__device__ __forceinline__ int argmax3_first(float a0, float a1, float a2) {
    // jnp.argmax semantics: first occurrence of the maximum.
    return (a0 >= a1 && a0 >= a2) ? 0 : ((a1 >= a2) ? 1 : 2);
}

__global__ __launch_bounds__(256) void slice_color_shader_kernel(
    const int*   __restrict__ pix_to_face,   // [NPIX]
    const float* __restrict__ bary,          // [NPIX, 3]
    const int*   __restrict__ faces,         // [F, 3]
    const f3*    __restrict__ verts_colors,  // [V] x 12B
    float*       __restrict__ out,           // [NPIX, 3]
    int npix)
{
    const int pixPerBlock = blockDim.x * 4;               // 1024
    const int blockBase   = blockIdx.x * pixPerBlock;     // scalar (uniform)
    const int base        = blockBase + threadIdx.x * 4;  // 4 px/thread

    // Block-uniform branch: scalar compare, no exec-mask manipulation.
    if (blockBase + pixPerBlock <= npix) {
        // ---- hot path: straight-line, fully vectorized 128-bit NT streams ----
        v4i f4 = __builtin_nontemporal_load((const v4i*)(pix_to_face + base));
        const float* bp = bary + (size_t)base * 3; // byte off mult. of 48 -> 16B aligned
        v4f w0 = __builtin_nontemporal_load((const v4f*)(bp + 0));
        v4f w1 = __builtin_nontemporal_load((const v4f*)(bp + 4));
        v4f w2 = __builtin_nontemporal_load((const v4f*)(bp + 8));

        const float bv[12] = { w0.x, w0.y, w0.z, w0.w,
                               w1.x, w1.y, w1.z, w1.w,
                               w2.x, w2.y, w2.z, w2.w };
        const int   fi[4]  = { f4.x, f4.y, f4.z, f4.w };
        float ov[12];

        #pragma unroll
        for (int p = 0; p < 4; ++p) {
            const int  idx = argmax3_first(bv[p*3 + 0], bv[p*3 + 1], bv[p*3 + 2]);
            const int  f   = fi[p];
            const bool m   = (f >= 0);
            const int  sf  = m ? f : 0;
            // L2-resident gathers (regular temporal hint, cached):
            const int  vrt = faces[sf * 3 + idx];        // global_load_b32
            const f3   c   = verts_colors[vrt];          // global_load_b96
            ov[p*3 + 0] = m ? c.x : 0.0f;
            ov[p*3 + 1] = m ? c.y : 0.0f;
            ov[p*3 + 2] = m ? c.z : 0.0f;
        }

        float* op = out + (size_t)base * 3;
        v4f s0 = { ov[0], ov[1], ov[2],  ov[3]  };
        v4f s1 = { ov[4], ov[5], ov[6],  ov[7]  };
        v4f s2 = { ov[8], ov[9], ov[10], ov[11] };
        __builtin_nontemporal_store(s0, (v4f*)(op + 0));
        __builtin_nontemporal_store(s1, (v4f*)(op + 4));
        __builtin_nontemporal_store(s2, (v4f*)(op + 8));
    } else {
        // ---- partial last block only (dead for npix = 8.39M); keep tiny ----
        #pragma clang loop unroll(disable)
        for (int p = base; p < npix && p < base + 4; ++p) {
            const float a0 = bary[(size_t)p * 3 + 0];
            const float a1 = bary[(size_t)p * 3 + 1];
            const float a2 = bary[(size_t)p * 3 + 2];
            const int   idx = argmax3_first(a0, a1, a2);
            const int   f   = pix_to_face[p];
            const bool  m   = (f >= 0);
            const int   sf  = m ? f : 0;
            const int   vrt = faces[sf * 3 + idx];
            const f3    c   = verts_colors[vrt];
            out[(size_t)p * 3 + 0] = m ? c.x : 0.0f;
            out[(size_t)p * 3 + 1] = m ? c.y : 0.0f;
            out[(size_t)p * 3 + 2] = m ? c.z : 0.0f;
        }
    }
}

extern "C" void kernel_launch(void* const* d_in, const int* in_sizes, int n_in,
                              void* d_out, int out_size, void* d_ws, size_t ws_size,
                              hipStream_t stream) {
    const int*   pix_to_face  = (const int*)  d_in[0];  // [B,H,W,1] int32
    const float* bary_coords  = (const float*)d_in[1];  // [B,H,W,1,3] f32
    const int*   faces        = (const int*)  d_in[2];  // [F,3] int32
    const f3*    verts_colors = (const f3*)   d_in[3];  // [V,3] f32
    float*       out          = (float*)d_out;          // [B,H,W,3] f32

    const int npix = in_sizes[0];                        // B*H*W = 8,388,608
    const int pixels_per_block = 256 * 4;                // 8 wave32s, 4 px/thread
    const int blocks = (npix + pixels_per_block - 1) / pixels_per_block;

    slice_color_shader_kernel<<<blocks, 256, 0, stream>>>(
        pix_to_face, bary_coords, faces, verts_colors, out, npix);
}